// Attention3_51453708206271
// MI455X (gfx1250) — compile-verified
//
#include <hip/hip_runtime.h>
#include <math.h>

typedef __attribute__((ext_vector_type(2))) float v2f;
typedef __attribute__((ext_vector_type(8))) float v8f;

#define B_    256
#define P_    128
#define K_    512
#define HE_   1024
#define HF2_  1024     // 2*HF
#define HID_  64
#define BT_STRIDE 68   // padded LDS stride -> conflict-free b64 B-frag reads
#define KCHUNK 64

// ---- CDNA5 async global->LDS copy (ASYNCcnt-tracked), builtin if available, else inline asm
#if defined(__has_builtin)
#if __has_builtin(__builtin_amdgcn_global_load_async_to_lds_b32)
#define HAVE_ASYNC_BUILTIN 1
#endif
#if __has_builtin(__builtin_amdgcn_s_wait_asynccnt)
#define HAVE_ASYNC_WAIT_BUILTIN 1
#endif
#endif

typedef __attribute__((address_space(1))) int as1_int;
typedef __attribute__((address_space(3))) int as3_int;

__device__ __forceinline__ void async_copy_b32(const float* g, float* l) {
#ifdef HAVE_ASYNC_BUILTIN
    __builtin_amdgcn_global_load_async_to_lds_b32(
        (as1_int*)g, (as3_int*)l, 0, 0);
#else
    unsigned lofs = (unsigned)(unsigned long long)l;   // low 32 bits = LDS offset
    asm volatile("global_load_async_to_lds_b32 %0, %1, off"
                 :: "v"(lofs), "v"((unsigned long long)g) : "memory");
#endif
}

__device__ __forceinline__ void async_wait0() {
#ifdef HAVE_ASYNC_WAIT_BUILTIN
    __builtin_amdgcn_s_wait_asynccnt(0);
#else
    asm volatile("s_wait_asynccnt 0x0" ::: "memory");
#endif
}

__launch_bounds__(256, 1)
__global__ void attn3_fused(const float* __restrict__ query,
                            const float* __restrict__ keys,
                            const float* __restrict__ frame,
                            const int*   __restrict__ mask,
                            const float* __restrict__ W1,
                            const float* __restrict__ W2,
                            float* __restrict__ out_emb,   // B x 1536
                            float* __restrict__ out_w)     // B x 128
{
    __shared__ __align__(16) float sBT[2][HID_ * BT_STRIDE]; // W1k chunk, transposed [n][kc], double-buffered
    __shared__ float sQF[HID_];
    __shared__ float sQPart[4][HID_];
    __shared__ float sScore[P_];
    __shared__ float sWgt[P_];
    __shared__ float sRed[256];

    const int b    = blockIdx.x;
    const int tid  = threadIdx.x;
    const int lane = tid & 31;
    const int wave = tid >> 5;
    const int lr   = lane & 15;   // M / N index within tile
    const int hi   = lane >> 4;   // K-pair selector for A/B frags

    // Per-thread staging coords: element idx = i*256+tid of a 64x64 chunk;
    // global reads contiguous, LDS destination transposed (per-lane scatter).
    const int stg_n  = tid & 63;
    const int stg_kc = tid >> 6;

    // ---------------- Prologue: kick off async DMA of W1k chunk 0 (overlaps Step 1)
    #pragma unroll
    for (int i = 0; i < (KCHUNK * HID_) / 256; ++i) {
        const int idx = i * 256 + tid;
        const int n   = stg_n;
        const int kc  = stg_kc + i * 4;        // idx>>6 = tid>>6 + i*4
        async_copy_b32(W1 + idx, &sBT[0][n * BT_STRIDE + kc]);
    }

    // ---------------- Step 1: qf[h] = query[b]·W1q[:,h] + frame[b]·W1f[:,h]
    {
        const int h = tid & 63;
        const int c = tid >> 6;                // 4 chunks of 512 j's
        const float* xq = query + (size_t)b * HE_;
        const float* xf = frame + (size_t)b * HF2_;
        float acc = 0.f;
        const int j0 = c * 512;
        #pragma unroll 4
        for (int jj = 0; jj < 512; ++jj) {
            const int j = j0 + jj;
            const float xv = (j < HE_) ? xq[j] : xf[j - HE_];
            acc = fmaf(xv, W1[(size_t)(K_ + j) * HID_ + h], acc);
        }
        sQPart[c][h] = acc;
    }
    __syncthreads();
    if (tid < HID_)
        sQF[tid] = sQPart[0][tid] + sQPart[1][tid] + sQPart[2][tid] + sQPart[3][tid];
    // sQF visibility covered by the barrier inside the chunk loop

    // ---------------- Step 2: h = keys[b] (128x512) @ W1k (512x64), WMMA f32,
    //                  double-buffered async LDS staging
    const int m0 = wave * 16;
    const float* keyRow = keys + ((size_t)b * P_ + (m0 + lr)) * K_;

    v8f acc0 = {}, acc1 = {}, acc2 = {}, acc3 = {};

    for (int c = 0; c < K_ / KCHUNK; ++c) {
        const int kk  = c * KCHUNK;
        const int cur = c & 1;

        async_wait0();       // my async writes into sBT[cur] have landed
        __syncthreads();     // everyone's writes landed; everyone done reading sBT[cur^1]

        if (c + 1 < K_ / KCHUNK) {
            const float* src = W1 + (size_t)(kk + KCHUNK) * HID_;
            float* dstBase = &sBT[cur ^ 1][stg_n * BT_STRIDE + stg_kc];
            #pragma unroll
            for (int i = 0; i < (KCHUNK * HID_) / 256; ++i)
                async_copy_b32(src + i * 256 + tid, dstBase + i * 4);
        }

        const float* bt = sBT[cur];
        #pragma unroll
        for (int ks = 0; ks < KCHUNK; ks += 4) {
            const int kc0 = ks + 2 * hi;          // local K of this lane's pair
            v2f A;
            {
                const float2 av = *(const float2*)(keyRow + kk + kc0);
                A.x = av.x; A.y = av.y;
            }
            v2f B0, B1, B2, B3;
            { float2 t = *(const float2*)(&bt[( 0 + lr) * BT_STRIDE + kc0]); B0.x = t.x; B0.y = t.y; }
            { float2 t = *(const float2*)(&bt[(16 + lr) * BT_STRIDE + kc0]); B1.x = t.x; B1.y = t.y; }
            { float2 t = *(const float2*)(&bt[(32 + lr) * BT_STRIDE + kc0]); B2.x = t.x; B2.y = t.y; }
            { float2 t = *(const float2*)(&bt[(48 + lr) * BT_STRIDE + kc0]); B3.x = t.x; B3.y = t.y; }
            acc0 = __builtin_amdgcn_wmma_f32_16x16x4_f32(false, A, false, B0, (short)0, acc0, false, false);
            acc1 = __builtin_amdgcn_wmma_f32_16x16x4_f32(false, A, false, B1, (short)0, acc1, false, false);
            acc2 = __builtin_amdgcn_wmma_f32_16x16x4_f32(false, A, false, B2, (short)0, acc2, false, false);
            acc3 = __builtin_amdgcn_wmma_f32_16x16x4_f32(false, A, false, B3, (short)0, acc3, false, false);
        }
    }

    // ---------------- Step 3: bias + relu + scores = h @ W2
    const float bias0 = sQF[ 0 + lr], bias1 = sQF[16 + lr];
    const float bias2 = sQF[32 + lr], bias3 = sQF[48 + lr];
    const float w20 = W2[ 0 + lr], w21 = W2[16 + lr];
    const float w22 = W2[32 + lr], w23 = W2[48 + lr];
    #pragma unroll
    for (int r = 0; r < 8; ++r) {
        float s = fmaxf(acc0[r] + bias0, 0.f) * w20
                + fmaxf(acc1[r] + bias1, 0.f) * w21
                + fmaxf(acc2[r] + bias2, 0.f) * w22
                + fmaxf(acc3[r] + bias3, 0.f) * w23;
        s += __shfl_xor(s, 8);
        s += __shfl_xor(s, 4);
        s += __shfl_xor(s, 2);
        s += __shfl_xor(s, 1);
        if (lr == 0) sScore[m0 + r + 8 * hi] = s;   // C/D layout: row = r + 8*hi
    }
    __syncthreads();

    // ---------------- Step 4: masked softmax over P=128
    float sval = -INFINITY;
    if (tid < P_)
        sval = (mask[(size_t)b * P_ + tid] != 0) ? sScore[tid] : -INFINITY;
    sRed[tid] = sval;                    // tid >= 128 padded with -inf
    __syncthreads();
    for (int st = 128; st > 0; st >>= 1) {
        if (tid < st) sRed[tid] = fmaxf(sRed[tid], sRed[tid + st]);
        __syncthreads();
    }
    const float mx = sRed[0];
    __syncthreads();
    const float ev = (tid < P_) ? expf(sval - mx) : 0.f;
    sRed[tid] = ev;
    __syncthreads();
    for (int st = 128; st > 0; st >>= 1) {
        if (tid < st) sRed[tid] += sRed[tid + st];
        __syncthreads();
    }
    const float denom = sRed[0];
    if (tid < P_) {
        const float w = ev / denom;
        sWgt[tid] = w;
        out_w[(size_t)b * P_ + tid] = w;
    }
    __syncthreads();

    // ---------------- Step 5: context = weights^T @ keys[b]; emit [ctx | frame]
    float* emb = out_emb + (size_t)b * (K_ + HF2_);
    for (int kb = 0; kb < K_; kb += 256) {
        const int k = kb + tid;
        const float* kcol = keys + (size_t)b * P_ * K_ + k;
        float ctx = 0.f;
        #pragma unroll 4
        for (int m = 0; m < P_; ++m)
            ctx = fmaf(sWgt[m], kcol[(size_t)m * K_], ctx);
        emb[k] = ctx;
    }
    const float* fr = frame + (size_t)b * HF2_;
    #pragma unroll
    for (int i = 0; i < HF2_ / 256; ++i)
        emb[K_ + i * 256 + tid] = fr[i * 256 + tid];
}

extern "C" void kernel_launch(void* const* d_in, const int* in_sizes, int n_in,
                              void* d_out, int out_size, void* d_ws, size_t ws_size,
                              hipStream_t stream) {
    const float* query = (const float*)d_in[0];
    const float* keys  = (const float*)d_in[1];
    const float* frame = (const float*)d_in[2];
    const int*   mask  = (const int*)  d_in[3];
    const float* W1    = (const float*)d_in[4];
    const float* W2    = (const float*)d_in[5];
    float* out_emb = (float*)d_out;
    float* out_w   = (float*)d_out + (size_t)B_ * (K_ + HF2_);
    attn3_fused<<<B_, 256, 0, stream>>>(query, keys, frame, mask, W1, W2, out_emb, out_w);
}